// DistLoss_41180146434814
// MI455X (gfx1250) — compile-verified
//
#include <hip/hip_runtime.h>
#include <hip/hip_fp16.h>

namespace {
constexpr int   kN   = 512;   // sequence length
constexpr int   kBC  = 192;   // 64 batches * 3 channels
constexpr float kEps = 1e-6f;
}

typedef float v2f __attribute__((ext_vector_type(2)));
typedef float v8f __attribute__((ext_vector_type(8)));

// One block per (b,c) sequence. Stage the 512-float row in LDS, then each of
// the 8 waves reduces 4 tiles of 16 shifts each (32 tiles cover s = 1..512,
// with s > 510 masked to zero). The inner sum-of-squares over j runs on the
// matrix pipe: D(16x16) = A(16x4) * ones(4x16) + C accumulates 16 row-sums.
// All masking is arithmetic (clamped index + 0/1 multiply) so the inner loop
// is branchless: EXEC stays all-ones and no exec-save/restore SALU appears.
__global__ void __launch_bounds__(256) dist_loss_partial(const float* __restrict__ x,
                                                         float* __restrict__ partials) {
  __shared__ float xs[kN];
  __shared__ float red[256];
  const int bc = blockIdx.x;    // b*3 + c
  const int t  = threadIdx.x;   // 0..255

  xs[t]       = x[bc * kN + t];
  xs[t + 256] = x[bc * kN + t + 256];
  __syncthreads();

  const int lane = t & 31;
  const int wave = t >> 5;      // 0..7
  const int m    = lane & 15;   // row (lag) within 16-lag tile
  const int h    = lane >> 4;   // half-wave selects which K pair this lane owns

  v2f bones; bones[0] = 1.0f; bones[1] = 1.0f;   // B = ones -> row sums

  float acc = 0.0f;             // replicated across columns; masked at write

  for (int tt = 0; tt < 4; ++tt) {
    const int   tile  = wave * 4 + tt;       // 0..31
    const int   s     = 1 + tile * 16 + m;   // this row's shift
    const float srowf = (s <= kN - 2) ? 1.0f : 0.0f;  // shifts are 1..510 only

    v8f c0 = {};                // two interleaved accumulators break the
    v8f c1 = {};                // WMMA->WMMA RAW chain on C

#pragma unroll 4
    for (int J0 = 0; J0 < kN; J0 += 8) {
      v2f a0, a1;
#pragma unroll
      for (int u = 0; u < 2; ++u) {          // u: which accumulator / K-chunk
        const int j0 = J0 + 4 * u + 2 * h;   // even, xs[j0], xs[j0+1] in-bounds
        const float x0 = xs[j0];
        const float x1 = xs[j0 + 1];
        const int jsa = j0 + s;              // shifted indices (may exceed 511)
        const int jsb = jsa + 1;
        const int jca = (jsa > kN - 1) ? (kN - 1) : jsa;   // clamp: in-bounds
        const int jcb = (jsb > kN - 1) ? (kN - 1) : jsb;
        const float y0 = xs[jca];            // unconditional LDS loads
        const float y1 = xs[jcb];
        const float m0 = (jsa < kN) ? srowf : 0.0f;        // 0/1 validity
        const float m1 = (jsb < kN) ? srowf : 0.0f;
        const float d0 = (x0 - y0 + kEps) * m0;            // (d*m)^2 == d^2*m
        const float d1 = (x1 - y1 + kEps) * m1;
        if (u == 0) { a0[0] = d0 * d0; a0[1] = d1 * d1; }
        else        { a1[0] = d0 * d0; a1[1] = d1 * d1; }
      }
      c0 = __builtin_amdgcn_wmma_f32_16x16x4_f32(
               false, a0, false, bones, (short)0, c0, false, false);
      c1 = __builtin_amdgcn_wmma_f32_16x16x4_f32(
               false, a1, false, bones, (short)0, c1, false, false);
    }

    // D columns are identical (B = ones). Lane layout: VGPR v of lane l holds
    // M = v + 8*(l>>4). Sum sqrt over the 8 lags this half-wave owns.
    float ssum = 0.0f;
#pragma unroll
    for (int v = 0; v < 8; ++v) {
      const float cv = c0[v] + c1[v];
      ssum += sqrtf(cv < 0.0f ? 0.0f : cv);
    }
    acc += ssum;
  }

  // Count each lag exactly once: only column-0 lanes (lane 0 and lane 16 of
  // each wave) contribute. Deterministic in-block serial reduction.
  red[t] = (m == 0) ? acc : 0.0f;
  __syncthreads();
  if (t == 0) {
    float ssum = 0.0f;
    for (int i = 0; i < 256; ++i) ssum += red[i];
    partials[bc] = ssum;
  }
}

// Deterministic fixed-order final reduction (no float atomics).
__global__ void dist_loss_final(const float* __restrict__ partials,
                                float* __restrict__ out) {
  if (threadIdx.x == 0) {
    float s = 0.0f;
    for (int i = 0; i < kBC; ++i) s += partials[i];
    out[0] = s * (1.0f / (3.0f * 510.0f * 64.0f));  // /C, /(N-2), /B
  }
}

extern "C" void kernel_launch(void* const* d_in, const int* in_sizes, int n_in,
                              void* d_out, int out_size, void* d_ws, size_t ws_size,
                              hipStream_t stream) {
  const float* x   = (const float*)d_in[0];
  float*       ws  = (float*)d_ws;   // 192 block partials
  float*       out = (float*)d_out;  // scalar
  dist_loss_partial<<<kBC, 256, 0, stream>>>(x, ws);
  dist_loss_final<<<1, 32, 0, stream>>>(ws, out);
}